// GenContactIsect_47691316854901
// MI455X (gfx1250) — compile-verified
//
#include <hip/hip_runtime.h>
#include <hip/hip_bf16.h>
#include <math.h>

#define N_PTS   10475
#define NFACES  1000
#define NMOUTH  30
#define NT16    655            // ceil(N_PTS/16)
#define NPAD    (NT16 * 16)    // 10480 (padded y/col count)
#define MBLK    164            // ceil(N_PTS/64)
#define MPAD    (MBLK * 64)    // 10496 (padded x/row count)
#define INF_BITS 0x7F800000u
#define BIGF    1e30f
#define THR2    (0.01f * 0.01f)

typedef __attribute__((ext_vector_type(2))) float v2f;
typedef __attribute__((ext_vector_type(8))) float v8f;
union V8 { v8f v; float f[8]; };

// ---------------- init workspace accumulators ----------------
__global__ void k_init(unsigned* colmin, unsigned* cnt) {
  int i = blockIdx.x * 256 + threadIdx.x;
  if (i < N_PTS) colmin[i] = INF_BITS;
  if (i == 0) *cnt = 0u;
}

// ---------------- close_mouth: mean of 30 verts (x and y) ----------------
__global__ void k_mouth(const float* x, const float* y, const int* vids,
                        float* mvx, float* mvy) {
  int t = threadIdx.x;
  if (t < 6) {
    const float* src = (t < 3) ? x : y;
    int c = t % 3;
    float s = 0.f;
    for (int k = 0; k < NMOUTH; ++k) s += src[vids[k] * 3 + c];
    float m = s / (float)NMOUTH;
    if (t < 3) mvx[c] = m; else mvy[c] = m;
  }
}

// -------- gather triangles + squared norms + padded y (stride-4 AoS) --------
__global__ void k_prep(const float* x, const float* y, const int* svid, const int* faces,
                       const float* mvx, const float* mvy,
                       float* trisX, float* trisY,
                       float* xxp, float* yp4, float* yyp) {
  int i = blockIdx.x * 256 + threadIdx.x;   // grid covers MPAD = 10496
  if (i < MPAD) {
    if (i < N_PTS) {
      float a0 = x[i*3], a1 = x[i*3+1], a2 = x[i*3+2];
      xxp[i] = a0*a0 + a1*a1 + a2*a2;
    } else {
      xxp[i] = BIGF;                        // pad rows -> huge distance
    }
  }
  if (i < NPAD) {
    if (i < N_PTS) {
      float b0 = y[i*3], b1 = y[i*3+1], b2 = y[i*3+2];
      yp4[i*4]   = b0; yp4[i*4+1] = b1; yp4[i*4+2] = b2; yp4[i*4+3] = 0.f;
      yyp[i] = b0*b0 + b1*b1 + b2*b2;
    } else {
      yp4[i*4] = 0.f; yp4[i*4+1] = 0.f; yp4[i*4+2] = 0.f; yp4[i*4+3] = 0.f;
      yyp[i] = BIGF;                        // pad cols -> huge distance
    }
  }
  if (i < NFACES * 3) {
    int f = i / 3, v = i % 3;
    int vid = svid[faces[f*3 + v]];         // index into closed mesh [0, N_PTS]
    #pragma unroll
    for (int c = 0; c < 3; ++c) {
      trisX[i*3 + c] = (vid == N_PTS) ? mvx[c] : x[vid*3 + c];
      trisY[i*3 + c] = (vid == N_PTS) ? mvy[c] : y[vid*3 + c];
    }
  }
}

// ---------------- winding numbers -> interior flags ----------------
__global__ void k_wind(const float* x, const float* y,
                       const float* trisX, const float* trisY,
                       unsigned* int1, unsigned* int2) {
  __shared__ float tr[NFACES * 9];          // 36 KB of 320 KB LDS
  int side = blockIdx.y;
  const float* pts   = side ? y : x;
  const float* tg    = side ? trisY : trisX;
  unsigned*    flags = side ? int2 : int1;
  for (int t = threadIdx.x; t < NFACES * 9; t += 256) tr[t] = tg[t];
  __syncthreads();
  int i = blockIdx.x * 256 + threadIdx.x;
  if (i >= N_PTS) return;
  float px = pts[i*3], py = pts[i*3+1], pz = pts[i*3+2];
  float acc = 0.f;
  for (int f = 0; f < NFACES; ++f) {
    const float* t9 = &tr[f * 9];
    float ax = t9[0]-px, ay = t9[1]-py, az = t9[2]-pz;
    float bx = t9[3]-px, by = t9[4]-py, bz = t9[5]-pz;
    float cx = t9[6]-px, cy = t9[7]-py, cz = t9[8]-pz;
    float la = sqrtf(ax*ax + ay*ay + az*az);
    float lb = sqrtf(bx*bx + by*by + bz*bz);
    float lc = sqrtf(cx*cx + cy*cy + cz*cz);
    float ux = by*cz - bz*cy;
    float uy = bz*cx - bx*cz;
    float uz = bx*cy - by*cx;
    float det = ax*ux + ay*uy + az*uz;
    float ab = ax*bx + ay*by + az*bz;
    float bc = bx*cx + by*cy + bz*cz;
    float ca = cx*ax + cy*ay + cz*az;
    float denom = la*lb*lc + ab*lc + bc*la + ca*lb;
    acc += 2.f * atan2f(det, denom);
  }
  float w = acc * 0.07957747154594767f;     // 1/(4*pi)
  flags[i] = (w >= 0.99f) ? 1u : 0u;
}

// ---------------- fused pairwise distance^2 via WMMA f32 16x16x4 ----------------
// d^2 = |x|^2 + |y|^2 - 2 x.y ; A pre-scaled by -2 so WMMA S = -2 x.y.
// sqrt is monotone -> track clamped d^2 mins, defer sqrt to the reductions.
// A layout (16x4 MxK): lane l: m=l%16, holds K = 2*(l/16)+{0,1}.
// B layout (4x16 KxN): lane l: n=l%16, VGPR v holds K = v + 2*(l/16).
// C/D layout (16x16):  lane l: n=l%16, VGPR v holds m = v + 8*(l/16).
__global__ void __launch_bounds__(256)
k_dist(const float* x, const float* yp4, const float* xxp, const float* yyp,
       float* rowmin_g, unsigned* colmin_g, unsigned* cnt_g) {
  __shared__ unsigned colmin_s[NPAD];       // ~42 KB (padded: unguarded atomics)
  __shared__ unsigned rowmin_s[64];
  __shared__ unsigned cnt_s;
  int tid = threadIdx.x;
  for (int t = tid; t < NPAD; t += 256) colmin_s[t] = INF_BITS;
  if (tid < 64) rowmin_s[tid] = INF_BITS;
  if (tid == 0) cnt_s = 0u;
  __syncthreads();

  int w = tid >> 5, lane = tid & 31;
  int half = lane >> 4, lm = lane & 15;
  int mt = w & 3, jstart = w >> 2;          // 2 waves share each m-tile
  int mbase = blockIdx.x * 64 + mt * 16;

  // A fragment (fixed per wave), scaled by -2; K=3 padded with 0
  int arow = mbase + lm;
  v2f a;
  if (arow < N_PTS) {
    int k0 = 2 * half;
    a.x = -2.f * x[arow*3 + k0];
    a.y = (k0 + 1 < 3) ? (-2.f * x[arow*3 + k0 + 1]) : 0.f;
  } else { a.x = 0.f; a.y = 0.f; }

  float lx2[8], rmin[8];
  #pragma unroll
  for (int v = 0; v < 8; ++v) {
    lx2[v]  = xxp[mbase + v + 8 * half];    // padded: unconditional
    rmin[v] = BIGF;
  }
  unsigned cnt = 0;
  const float2* yp2 = (const float2*)yp4;   // [n*2 + half] = (y[k0], y[k0+1]/0)

  for (int j = jstart; j < NT16; j += 2) {  // wave-uniform: EXEC all-1s at WMMA
    int n = j * 16 + lm;
    float2 t = yp2[n * 2 + half];           // one aligned b64 load
    v2f b; b.x = t.x; b.y = t.y;
    float ly2 = yyp[n];

    v8f c = {0.f, 0.f, 0.f, 0.f, 0.f, 0.f, 0.f, 0.f};
    V8 s;
    s.v = __builtin_amdgcn_wmma_f32_16x16x4_f32(false, a, false, b,
                                                (short)0, c, false, false);
    float cmin = BIGF;
    #pragma unroll
    for (int v = 0; v < 8; ++v) {
      float d2 = fmaxf(lx2[v] + ly2 + s.f[v], 0.f);
      cnt += (d2 < THR2) ? 1u : 0u;         // d < 0.01  <=>  d^2 < 0.01^2
      rmin[v] = fminf(rmin[v], d2);
      cmin    = fminf(cmin, d2);
    }
    atomicMin(&colmin_s[n], __float_as_uint(cmin));   // padded: no guard
  }

  #pragma unroll
  for (int v = 0; v < 8; ++v) {
    int local = mt * 16 + v + 8 * half;     // block-local row in [0,64)
    atomicMin(&rowmin_s[local], __float_as_uint(rmin[v]));
  }
  atomicAdd(&cnt_s, cnt);
  __syncthreads();

  if (tid < 64) {
    int m = blockIdx.x * 64 + tid;
    if (m < N_PTS)                          // rows block-exclusive: plain store
      rowmin_g[m] = sqrtf(__uint_as_float(rowmin_s[tid]));
  }
  for (int t = tid; t < N_PTS; t += 256) atomicMin(&colmin_g[t], colmin_s[t]);
  if (tid == 0) atomicAdd(cnt_g, cnt_s);
}

// ---------------- masked stats: max / mean / exact median (radix select) ----------------
__global__ void k_stats(const float* rowmin, const unsigned* colmin_b,
                        const unsigned* int1, const unsigned* int2,
                        const unsigned* cnt_g, float* out) {
  __shared__ float    red_f[256];
  __shared__ unsigned red_u[256];
  __shared__ unsigned hist[256];
  __shared__ unsigned sh_prefix, sh_k;
  int tid = threadIdx.x;

  for (int side = 0; side < 2; ++side) {
    const unsigned* mask = side ? int2 : int1;
    unsigned c = 0; float sum = 0.f; float mx = -BIGF;
    for (int i = tid; i < N_PTS; i += 256) {
      float v = side ? sqrtf(__uint_as_float(colmin_b[i])) : rowmin[i];
      if (mask[i]) { c++; sum += v; mx = fmaxf(mx, v); }
    }
    red_u[tid] = c; red_f[tid] = sum;
    __syncthreads();
    for (int s = 128; s > 0; s >>= 1) {
      if (tid < s) { red_u[tid] += red_u[tid + s]; red_f[tid] += red_f[tid + s]; }
      __syncthreads();
    }
    unsigned cnt = red_u[0]; float total = red_f[0];
    __syncthreads();
    red_f[tid] = mx;
    __syncthreads();
    for (int s = 128; s > 0; s >>= 1) {
      if (tid < s) red_f[tid] = fmaxf(red_f[tid], red_f[tid + s]);
      __syncthreads();
    }
    float maxv = red_f[0];
    __syncthreads();

    // exact k-th smallest, k=(cnt-1)/2, over keys (nonneg floats -> monotone bits)
    unsigned k = (cnt > 0) ? ((cnt - 1u) >> 1) : 0u;
    unsigned prefix = 0u, prefmask = 0u;
    for (int pass = 0; pass < 4; ++pass) {
      int shift = 24 - 8 * pass;
      hist[tid] = 0u;
      __syncthreads();
      for (int i = tid; i < N_PTS; i += 256) {
        float v = side ? sqrtf(__uint_as_float(colmin_b[i])) : rowmin[i];
        unsigned u = mask[i] ? __float_as_uint(v) : INF_BITS;
        if ((u & prefmask) == prefix) atomicAdd(&hist[(u >> shift) & 255u], 1u);
      }
      __syncthreads();
      if (tid == 0) {
        unsigned kk = k, dg;
        for (dg = 0; dg < 255u; ++dg) {
          unsigned h = hist[dg];
          if (kk < h) break;
          kk -= h;
        }
        sh_prefix = prefix | (dg << shift);
        sh_k = kk;
      }
      __syncthreads();
      prefix = sh_prefix; k = sh_k;
      prefmask |= (0xFFu << shift);
      __syncthreads();
    }
    if (tid == 0) {
      out[1 + 3*side] = (cnt > 0) ? maxv : 0.f;
      out[2 + 3*side] = (cnt > 0) ? (total / (float)cnt) : 0.f;
      out[3 + 3*side] = (cnt > 0) ? __uint_as_float(prefix) : 0.f;
    }
    __syncthreads();
  }
  if (tid == 0) out[0] = (float)(*cnt_g);
}

// ---------------- host launcher ----------------
extern "C" void kernel_launch(void* const* d_in, const int* in_sizes, int n_in,
                              void* d_out, int out_size, void* d_ws, size_t ws_size,
                              hipStream_t stream) {
  const float* x     = (const float*)d_in[0];
  const float* y     = (const float*)d_in[1];
  const int*   vids  = (const int*)d_in[2];
  const int*   svid  = (const int*)d_in[3];
  const int*   faces = (const int*)d_in[4];
  float* out = (float*)d_out;
  float* ws  = (float*)d_ws;

  float*    trisX  = ws;                          // 9000
  float*    trisY  = trisX + 9000;                // 9000
  float*    yp4    = trisY + 9000;                // NPAD*4  (16B-aligned: offset 18000 floats = 72000 B)
  float*    yyp    = yp4 + NPAD * 4;              // NPAD
  float*    xxp    = yyp + NPAD;                  // MPAD
  float*    rowmin = xxp + MPAD;                  // N_PTS
  unsigned* colmin = (unsigned*)(rowmin + N_PTS); // N_PTS
  unsigned* int1   = colmin + N_PTS;              // N_PTS
  unsigned* int2   = int1 + N_PTS;                // N_PTS
  unsigned* cntg   = int2 + N_PTS;                // 1
  float*    mvx    = (float*)(cntg + 1);          // 3
  float*    mvy    = mvx + 3;                     // 3   (total ~492 KB)

  k_init <<<(N_PTS + 255) / 256, 256, 0, stream>>>(colmin, cntg);
  k_mouth<<<1, 64, 0, stream>>>(x, y, vids, mvx, mvy);
  k_prep <<<(MPAD + 255) / 256, 256, 0, stream>>>(x, y, svid, faces, mvx, mvy,
                                                  trisX, trisY, xxp, yp4, yyp);
  dim3 gw((N_PTS + 255) / 256, 2);
  k_wind <<<gw, 256, 0, stream>>>(x, y, trisX, trisY, int1, int2);
  k_dist <<<MBLK, 256, 0, stream>>>(x, yp4, xxp, yyp, rowmin, colmin, cntg);
  k_stats<<<1, 256, 0, stream>>>(rowmin, colmin, int1, int2, cntg, out);
}